// GCN_7481833030017
// MI455X (gfx1250) — compile-verified
//
#include <hip/hip_runtime.h>
#include <hip/hip_bf16.h>

typedef __attribute__((ext_vector_type(2))) float v2f;
typedef __attribute__((ext_vector_type(8))) float v8f;

#define NCLS 10

__device__ __forceinline__ void atomAddF(float* p, float v) {
    unsafeAtomicAdd(p, v);   // native global_atomic_add_f32 (L2-resident buffers)
}

// ---------- utility kernels ----------
__global__ void zero_f32(float* __restrict__ p, long n) {
    long i = (long)blockIdx.x * blockDim.x + threadIdx.x;
    if (i < n) p[i] = 0.f;
}

__global__ void set_f32(float* __restrict__ p, float v, int n) {
    int i = blockIdx.x * blockDim.x + threadIdx.x;
    if (i < n) p[i] = v;
}

// ---------- degree / normalization ----------
__global__ void deg_accum(const int* __restrict__ dst, float* __restrict__ deg, int E) {
    int e = blockIdx.x * blockDim.x + threadIdx.x;
    if (e < E) atomAddF(&deg[dst[e]], 1.0f);
}

__global__ void rsqrt_k(const float* __restrict__ deg, float* __restrict__ dinv, int n) {
    int i = blockIdx.x * blockDim.x + threadIdx.x;
    if (i < n) dinv[i] = rsqrtf(deg[i]);
}

// ---------- f32 WMMA GEMM: C[N,M] = A[N,K] @ W[K,M] ----------
// Block = 8 waves, each wave owns a 16-row tile and CT 16-col tiles (CT accumulators).
// W staged in LDS transposed+zero-padded: lds[col*Kstride + k], col < CT*16, k < Kstride.
//   - K rows >= K are zero  -> A may read garbage for padded k (clamped index, x0 = 0)
//   - cols >= M are zero    -> lanes for padded cols compute zeros, never stored
// A frag (16x4 f32): lanes 0-15 hold K=k0,k0+1 of row tm+l; lanes 16-31 hold K=k0+2,k0+3.
//   abase+ka is always even (K,ka even) -> 8B-aligned v2f load (global_load_b64).
// D tile: vgpr j -> row = tm + 8*half + j, col = 16*ct + (lane&15).
template <int CT>
__global__ __launch_bounds__(256) void wmma_gemm_f32(
        const float* __restrict__ A, const float* __restrict__ W,
        float* __restrict__ C, int N, int K, int Kpad, int Kstride, int M) {
    extern __shared__ float lds[];
    const int Mpad = CT * 16;

    // cooperative staged load of W (transposed, zero-padded)
    for (int i = threadIdx.x; i < Mpad * Kstride; i += 256) {
        int col = i / Kstride;
        int kk  = i - col * Kstride;
        lds[i] = (kk < K && col < M) ? W[kk * M + col] : 0.f;
    }
    __syncthreads();

    int wave = threadIdx.x >> 5;
    int lane = threadIdx.x & 31;
    int half = lane >> 4;
    int l    = lane & 15;
    int tm   = (blockIdx.x * 8 + wave) << 4;
    if (tm >= N) return;

    const int NK    = N * K;
    const int arow  = min(tm + l, N - 1);
    const int abase = arow * K;

    v8f acc[CT];
#pragma unroll
    for (int ct = 0; ct < CT; ++ct) acc[ct] = (v8f){};

    for (int k0 = 0; k0 < Kpad; k0 += 4) {
        int ka = k0 + half * 2;
        // single 8B A-fragment load (clamped pair base; garbage x zero-padded W = 0)
        v2f a = *(const v2f*)&A[min(abase + ka, NK - 2)];
        // batch all B fragments first so loads clause + single wait,
        // then CT back-to-back WMMAs
        v2f b[CT];
#pragma unroll
        for (int ct = 0; ct < CT; ++ct)
            b[ct] = *(const v2f*)&lds[(ct * 16 + l) * Kstride + ka];
#pragma unroll
        for (int ct = 0; ct < CT; ++ct)
            acc[ct] = __builtin_amdgcn_wmma_f32_16x16x4_f32(
                false, a, false, b[ct], (short)0, acc[ct], false, false);
    }

#pragma unroll
    for (int ct = 0; ct < CT; ++ct) {
        int col = ct * 16 + l;
        if (col >= M) continue;
#pragma unroll
        for (int j = 0; j < 8; ++j) {
            int row = tm + half * 8 + j;
            if (row < N) C[row * M + col] = acc[ct][j];
        }
    }
}

// ---------- sparse aggregation: agg[dst] += h[src] * dinv[src]*dinv[dst] ----------
template <int F>
__global__ void edge_agg(const int* __restrict__ src, const int* __restrict__ dst,
                         const float* __restrict__ dinv, const float* __restrict__ h,
                         float* __restrict__ agg, int E) {
    int tid = blockIdx.x * blockDim.x + threadIdx.x;
    if (tid >= E * F) return;
    int e = tid / F, f = tid - e * F;
    int s = src[e], d = dst[e];
    float norm = dinv[s] * dinv[d];
    atomAddF(&agg[d * F + f], h[s * F + f] * norm);
}

// ---------- self-loop + bias: agg[i,f] += h[i,f]/deg[i] + b[f] ----------
__global__ void selfloop_bias(float* __restrict__ agg, const float* __restrict__ h,
                              const float* __restrict__ deg, const float* __restrict__ bias,
                              int N, int F) {
    long tid = (long)blockIdx.x * blockDim.x + threadIdx.x;
    if (tid >= (long)N * F) return;
    int i = (int)(tid / F);
    int f = (int)(tid - (long)i * F);
    agg[tid] += h[tid] / deg[i] + bias[f];
}

// ---------- BN column stats (sum, sumsq) with LDS reduction; F divides 256 ----------
__global__ __launch_bounds__(256) void col_stats(const float* __restrict__ y,
                                                 float* __restrict__ colsum,
                                                 float* __restrict__ colsq, int N, int F) {
    __shared__ float ssum[256], ssq[256];
    int t = threadIdx.x;
    int rpb = 256 / F;
    int col = t % F;
    int roff = t / F;
    float s = 0.f, q = 0.f;
    for (int r = blockIdx.x * rpb + roff; r < N; r += gridDim.x * rpb) {
        float v = y[(long)r * F + col];
        s += v; q += v * v;
    }
    ssum[t] = s; ssq[t] = q;
    __syncthreads();
    if (t < F) {
        for (int k = t + F; k < 256; k += F) { s += ssum[k]; q += ssq[k]; }
        atomAddF(&colsum[col], s);
        atomAddF(&colsq[col], q);
    }
}

__global__ void bn_finalize(const float* __restrict__ colsum, const float* __restrict__ colsq,
                            const float* __restrict__ g, const float* __restrict__ be,
                            float* __restrict__ scale, float* __restrict__ shift,
                            float invN, int F) {
    int f = blockIdx.x * blockDim.x + threadIdx.x;
    if (f < F) {
        float mu = colsum[f] * invN;
        float var = colsq[f] * invN - mu * mu;
        float sc = g[f] * rsqrtf(var + 1e-5f);
        scale[f] = sc;
        shift[f] = be[f] - mu * sc;
    }
}

__global__ void bn_apply_relu(float* __restrict__ y, const float* __restrict__ scale,
                              const float* __restrict__ shift, long n, int F) {
    long tid = (long)blockIdx.x * blockDim.x + threadIdx.x;
    if (tid >= n) return;
    int f = (int)(tid % F);
    y[tid] = fmaxf(0.f, y[tid] * scale[f] + shift[f]);
}

// ---------- global mean pool + log_softmax ----------
__global__ void pool_accum(const float* __restrict__ h, const int* __restrict__ batch,
                           float* __restrict__ psum, float* __restrict__ pcnt, int N) {
    int tid = blockIdx.x * blockDim.x + threadIdx.x;
    if (tid >= N * NCLS) return;
    int i = tid / NCLS, f = tid - i * NCLS;
    int g = batch[i];
    atomAddF(&psum[g * NCLS + f], h[tid]);
    if (f == 0) atomAddF(&pcnt[g], 1.f);
}

__global__ void pool_logsoftmax(const float* __restrict__ psum, const float* __restrict__ pcnt,
                                float* __restrict__ out, int G) {
    int g = blockIdx.x * blockDim.x + threadIdx.x;
    if (g >= G) return;
    float c = fmaxf(pcnt[g], 1.0f);
    float v[NCLS], m = -3.4e38f;
#pragma unroll
    for (int f = 0; f < NCLS; ++f) { v[f] = psum[g * NCLS + f] / c; m = fmaxf(m, v[f]); }
    float lse = 0.f;
#pragma unroll
    for (int f = 0; f < NCLS; ++f) lse += expf(v[f] - m);
    lse = logf(lse);
#pragma unroll
    for (int f = 0; f < NCLS; ++f) out[g * NCLS + f] = v[f] - m - lse;
}

// ---------------------------------------------------------------------------
extern "C" void kernel_launch(void* const* d_in, const int* in_sizes, int n_in,
                              void* d_out, int out_size, void* d_ws, size_t ws_size,
                              hipStream_t stream) {
    const float* x   = (const float*)d_in[0];
    const int*   ei  = (const int*)d_in[1];
    const int*   bat = (const int*)d_in[2];
    const float* W1 = (const float*)d_in[3],  *b1 = (const float*)d_in[4];
    const float* g1 = (const float*)d_in[5],  *be1 = (const float*)d_in[6];
    const float* W2 = (const float*)d_in[7],  *b2 = (const float*)d_in[8];
    const float* g2 = (const float*)d_in[9],  *be2 = (const float*)d_in[10];
    const float* W3 = (const float*)d_in[11], *b3 = (const float*)d_in[12];
    float* out = (float*)d_out;

    const int N  = in_sizes[2];
    const int E  = in_sizes[1] / 2;
    const int F0 = in_sizes[0] / N;          // 6
    const int G  = out_size / NCLS;          // 1000
    const int* src = ei;
    const int* dst = ei + E;

    // workspace layout
    char* ws = (char*)d_ws;
    size_t off = 0;
    auto alloc = [&](size_t bytes) { size_t o = off; off = (off + bytes + 255) & ~(size_t)255; return o; };
    float* deg    = (float*)(ws + alloc((size_t)N * 4));
    float* dinv   = (float*)(ws + alloc((size_t)N * 4));
    float* colsum = (float*)(ws + alloc(512));
    float* colsq  = (float*)(ws + alloc(512));
    float* scale  = (float*)(ws + alloc(512));
    float* shift  = (float*)(ws + alloc(512));
    float* psum   = (float*)(ws + alloc((size_t)G * NCLS * 4));
    float* pcnt   = (float*)(ws + alloc((size_t)G * 4));
    float* bufH   = (float*)(ws + alloc((size_t)N * 128 * 4));  // h = x@W
    float* bufA   = (float*)(ws + alloc((size_t)N * 128 * 4));  // agg / activations

    auto blocks1d = [](long n) { return (unsigned)((n + 255) / 256); };

    // degrees (with self-loop) and deg^{-1/2}, shared by all layers
    set_f32<<<blocks1d(N), 256, 0, stream>>>(deg, 1.0f, N);
    deg_accum<<<blocks1d(E), 256, 0, stream>>>(dst, deg, E);
    rsqrt_k<<<blocks1d(N), 256, 0, stream>>>(deg, dinv, N);

    auto gemm = [&](const float* A, const float* W, float* C, int K, int M) {
        int Kpad    = (K + 3) & ~3;
        int Kstride = Kpad + 2;              // dodge LDS bank-stride conflicts
        int blocks  = (N + 127) / 128;       // 8 row-tiles per block
        int ct      = (M + 15) / 16;
        size_t ldsB = (size_t)(ct * 16) * Kstride * 4;
        if (ct == 8)      wmma_gemm_f32<8><<<blocks, 256, ldsB, stream>>>(A, W, C, N, K, Kpad, Kstride, M);
        else if (ct == 4) wmma_gemm_f32<4><<<blocks, 256, ldsB, stream>>>(A, W, C, N, K, Kpad, Kstride, M);
        else              wmma_gemm_f32<1><<<blocks, 256, ldsB, stream>>>(A, W, C, N, K, Kpad, Kstride, M);
    };
    auto aggregate = [&](const float* b, int M) {
        zero_f32<<<blocks1d((long)N * M), 256, 0, stream>>>(bufA, (long)N * M);
        if (M == 64)       edge_agg<64> <<<blocks1d((long)E * 64),  256, 0, stream>>>(src, dst, dinv, bufH, bufA, E);
        else if (M == 128) edge_agg<128><<<blocks1d((long)E * 128), 256, 0, stream>>>(src, dst, dinv, bufH, bufA, E);
        else               edge_agg<NCLS><<<blocks1d((long)E * NCLS), 256, 0, stream>>>(src, dst, dinv, bufH, bufA, E);
        selfloop_bias<<<blocks1d((long)N * M), 256, 0, stream>>>(bufA, bufH, deg, b, N, M);
    };
    auto layer = [&](const float* Ain, const float* W, const float* b, int K, int M) {
        gemm(Ain, W, bufH, K, M);            // h = Ain @ W  (WMMA)
        aggregate(b, M);
    };
    auto bn_relu = [&](const float* g, const float* be, int M) {
        zero_f32<<<1, 256, 0, stream>>>(colsum, M);
        zero_f32<<<1, 256, 0, stream>>>(colsq, M);
        col_stats<<<512, 256, 0, stream>>>(bufA, colsum, colsq, N, M);
        bn_finalize<<<1, 256, 0, stream>>>(colsum, colsq, g, be, scale, shift, 1.0f / (float)N, M);
        bn_apply_relu<<<blocks1d((long)N * M), 256, 0, stream>>>(bufA, scale, shift, (long)N * M, M);
    };

    // layer 1: x[N,6] -> 64, BN, relu
    layer(x, W1, b1, F0, 64);
    bn_relu(g1, be1, 64);
    // layer 2: 64 -> 128, BN, relu
    layer(bufA, W2, b2, 64, 128);
    bn_relu(g2, be2, 128);
    // layer 3: 128 -> 10
    layer(bufA, W3, b3, 128, NCLS);

    // global mean pool + log_softmax
    zero_f32<<<blocks1d((long)G * NCLS), 256, 0, stream>>>(psum, (long)G * NCLS);
    zero_f32<<<blocks1d(G), 256, 0, stream>>>(pcnt, G);
    pool_accum<<<blocks1d((long)N * NCLS), 256, 0, stream>>>(bufA, bat, psum, pcnt, N);
    pool_logsoftmax<<<blocks1d(G), 256, 0, stream>>>(psum, pcnt, out, G);
}